// Prompt_36447092474476
// MI455X (gfx1250) — compile-verified
//
#include <hip/hip_runtime.h>
#include <hip/hip_bf16.h>
#include <math.h>

// ---------------------------------------------------------------------------
// L2P prompt selection:
//   x_embed (32,2048,768) f32, prompt (1024,5,768) f32, prompt_key (1024,768) f32
// Outputs (flat, concatenated):
//   prompted_embedding (32,2053,768) | sim (32,1024) | top_k_sim (32,5)
//   | top_k_idx (32,5) [as float] | idx (32) [as float]
// ---------------------------------------------------------------------------

typedef __attribute__((ext_vector_type(2))) float v2f;
typedef __attribute__((ext_vector_type(8))) float v8f;

#define B_DIM   32
#define N_DIM   2048
#define D_DIM   768
#define D4      (D_DIM / 4)      // 192 float4 per row
#define POOL    1024
#define PLEN    5
#define OUT_N   (N_DIM + PLEN)   // 2053
#define NSEG    16
#define SEGROWS (N_DIM / NSEG)   // 128
#define TOPK    5

// ---------------------------------------------------------------------------
// Kernel 1: fused copy x_embed -> out[:, 5:, :]  AND  segmented partial sums
// over N. One pass over the 201 MB tensor (the memory-bound hot loop).
// grid = 32*16 blocks, block = 192 threads (6 waves), float4 per thread.
// ---------------------------------------------------------------------------
__global__ void k_copy_partial(const float* __restrict__ xe,
                               float* __restrict__ out,
                               float* __restrict__ partial) {
    const int blk = blockIdx.x;
    const int b   = blk >> 4;      // 0..31
    const int seg = blk & 15;      // 0..15
    const int t   = threadIdx.x;   // 0..191  (float4 column)

    const float4* __restrict__ src =
        (const float4*)(xe + ((size_t)b * N_DIM + (size_t)seg * SEGROWS) * D_DIM);
    float4* __restrict__ dst =
        (float4*)(out + ((size_t)b * OUT_N + PLEN + (size_t)seg * SEGROWS) * D_DIM);

    float4 acc; acc.x = 0.f; acc.y = 0.f; acc.z = 0.f; acc.w = 0.f;
    for (int r = 0; r < SEGROWS; ++r) {
        float4 v = src[(size_t)r * D4 + t];
        dst[(size_t)r * D4 + t] = v;
        acc.x += v.x; acc.y += v.y; acc.z += v.z; acc.w += v.w;
    }
    ((float4*)partial)[((size_t)b * NSEG + seg) * D4 + t] = acc;
}

// ---------------------------------------------------------------------------
// Kernel 2: reduce the 16 partials -> x_sum[b][d]; also invx[b]=rsqrt(||x||^2).
// (Normalization is scale invariant, so we use the raw sum, not the mean.)
// grid = 32 blocks, block = 192 threads.
// ---------------------------------------------------------------------------
__global__ void k_xsum(const float* __restrict__ partial,
                       float* __restrict__ xsum,
                       float* __restrict__ invx) {
    const int b = blockIdx.x;
    const int t = threadIdx.x;   // 0..191

    const float4* __restrict__ p = ((const float4*)partial) + (size_t)b * NSEG * D4;
    float4 acc; acc.x = 0.f; acc.y = 0.f; acc.z = 0.f; acc.w = 0.f;
    for (int s = 0; s < NSEG; ++s) {
        float4 v = p[(size_t)s * D4 + t];
        acc.x += v.x; acc.y += v.y; acc.z += v.z; acc.w += v.w;
    }
    ((float4*)xsum)[(size_t)b * D4 + t] = acc;

    __shared__ float red[256];
    red[t] = acc.x * acc.x + acc.y * acc.y + acc.z * acc.z + acc.w * acc.w;
    if (t < 64) red[192 + t] = 0.f;
    __syncthreads();
    for (int ofs = 128; ofs > 0; ofs >>= 1) {
        if (t < ofs) red[t] += red[t + ofs];
        __syncthreads();
    }
    if (t == 0) invx[b] = rsqrtf(fmaxf(red[0], 1e-12f));
}

// ---------------------------------------------------------------------------
// Kernel 3: per-prompt-key inverse L2 norms.  grid = 1024, block = 192.
// ---------------------------------------------------------------------------
__global__ void k_invp(const float* __restrict__ pk,
                       float* __restrict__ invp) {
    const int p = blockIdx.x;
    const int t = threadIdx.x;
    float4 v = ((const float4*)pk)[(size_t)p * D4 + t];

    __shared__ float red[256];
    red[t] = v.x * v.x + v.y * v.y + v.z * v.z + v.w * v.w;
    if (t < 64) red[192 + t] = 0.f;
    __syncthreads();
    for (int ofs = 128; ofs > 0; ofs >>= 1) {
        if (t < ofs) red[t] += red[t + ofs];
        __syncthreads();
    }
    if (t == 0) invp[p] = rsqrtf(fmaxf(red[0], 1e-12f));
}

// ---------------------------------------------------------------------------
// Kernel 4: sim = (x_sum . pk^T) * invx * invp via V_WMMA_F32_16X16X4_F32.
// One wave (32 threads) per 16x16 output tile; 2 x 64 = 128 tiles.
// A-layout (f32 16x4): lanes 0-15 carry K={k0,k0+1}, lanes 16-31 K={k0+2,k0+3}
// for row M=lane&15.  B (4x16) is the transpose analog: lane halves carry the
// K row pairs for column N=lane&15.  EXEC is all ones (flat wave, no branches).
// ---------------------------------------------------------------------------
__global__ void k_sim_wmma(const float* __restrict__ xsum,
                           const float* __restrict__ pk,
                           const float* __restrict__ invx,
                           const float* __restrict__ invp,
                           float* __restrict__ sim) {
    const int tile  = blockIdx.x;   // 0..127
    const int mtile = tile & 1;     // 0..1   (M base = mtile*16)
    const int ntile = tile >> 1;    // 0..63  (N base = ntile*16)
    const int lane  = threadIdx.x;  // 0..31
    const int lo    = lane & 15;
    const int hi    = lane >> 4;

    const int m = mtile * 16 + lo;
    const int n = ntile * 16 + lo;
    const float* __restrict__ arow = xsum + (size_t)m * D_DIM;
    const float* __restrict__ brow = pk   + (size_t)n * D_DIM;

    v8f c = {};
    for (int k = 0; k < D_DIM; k += 4) {
        const int kk = k + 2 * hi;                 // even -> 8B aligned float2
        v2f a = *(const v2f*)(arow + kk);
        v2f b = *(const v2f*)(brow + kk);
        // (neg_a, A, neg_b, B, c_mod, C, reuse_a, reuse_b)
        c = __builtin_amdgcn_wmma_f32_16x16x4_f32(false, a, false, b,
                                                  (short)0, c, false, false);
    }

    // C/D layout: element r of the v8f at this lane is (M = r + 8*hi, N = n).
    const float ipn = invp[n];
    #pragma unroll
    for (int r = 0; r < 8; ++r) {
        const int mm = mtile * 16 + r + 8 * hi;
        sim[(size_t)mm * POOL + n] = c[r] * invx[mm] * ipn;
    }
}

// ---------------------------------------------------------------------------
// Kernel 5: top-5 per row, descending, ties -> lower index (jax.lax.top_k).
// One wave per batch row: lane-local scan (stride 32) + shfl_xor argmax merge,
// 5 rounds with exclusion of already-chosen indices.
// ---------------------------------------------------------------------------
__global__ void k_topk(const float* __restrict__ sim,
                       float* __restrict__ tk_sim,
                       float* __restrict__ tk_idx,
                       float* __restrict__ idx_f,
                       int* __restrict__ idx_i) {
    const int b    = blockIdx.x;
    const int lane = threadIdx.x;   // 0..31
    const float* __restrict__ row = sim + (size_t)b * POOL;

    int chosen[TOPK];
    #pragma unroll
    for (int j = 0; j < TOPK; ++j) chosen[j] = -1;

    for (int j = 0; j < TOPK; ++j) {
        float bv = -3.402823466e38f;
        int   bi = 0x7FFFFFFF;
        for (int e = lane; e < POOL; e += 32) {
            bool skip = false;
            #pragma unroll
            for (int q = 0; q < TOPK; ++q) skip = skip || (q < j && chosen[q] == e);
            if (skip) continue;
            const float v = row[e];
            if (v > bv || (v == bv && e < bi)) { bv = v; bi = e; }
        }
        #pragma unroll
        for (int ofs = 16; ofs > 0; ofs >>= 1) {
            const float ov = __shfl_xor(bv, ofs, 32);
            const int   oi = __shfl_xor(bi, ofs, 32);
            if (ov > bv || (ov == bv && oi < bi)) { bv = ov; bi = oi; }
        }
        chosen[j] = bi;  // identical on all lanes after full reduction
        if (lane == 0) {
            tk_sim[b * TOPK + j] = bv;
            tk_idx[b * TOPK + j] = (float)bi;
            if (j == 0) { idx_f[b] = (float)bi; idx_i[b] = bi; }
        }
    }
}

// ---------------------------------------------------------------------------
// Kernel 6: out[b][0..4][:] = prompt[idx[b]][:][:]   (float4 copies)
// 32*5*192 = 30720 float4 -> 120 blocks x 256 threads.
// ---------------------------------------------------------------------------
__global__ void k_gather(const float* __restrict__ prompt,
                         const int* __restrict__ idx_i,
                         float* __restrict__ out) {
    const int t = blockIdx.x * blockDim.x + threadIdx.x;
    const int total = B_DIM * PLEN * D4;
    if (t >= total) return;
    const int d4  = t % D4;
    const int rem = t / D4;
    const int r   = rem % PLEN;
    const int b   = rem / PLEN;
    const int pi  = idx_i[b];
    float4 v = ((const float4*)prompt)[((size_t)pi * PLEN + r) * D4 + d4];
    ((float4*)out)[((size_t)b * OUT_N + r) * D4 + d4] = v;
}

// ---------------------------------------------------------------------------
extern "C" void kernel_launch(void* const* d_in, const int* in_sizes, int n_in,
                              void* d_out, int out_size, void* d_ws, size_t ws_size,
                              hipStream_t stream) {
    (void)in_sizes; (void)n_in; (void)out_size; (void)ws_size;

    const float* xe     = (const float*)d_in[0];   // (32,2048,768)
    const float* prompt = (const float*)d_in[1];   // (1024,5,768)
    const float* pk     = (const float*)d_in[2];   // (1024,768)

    float* out = (float*)d_out;
    float* pe  = out;                                              // (32,2053,768)
    float* sim = out + (size_t)B_DIM * OUT_N * D_DIM;              // (32,1024)
    float* tks = sim + (size_t)B_DIM * POOL;                       // (32,5)
    float* tki = tks + B_DIM * TOPK;                               // (32,5)
    float* idf = tki + B_DIM * TOPK;                               // (32,)

    float* ws      = (float*)d_ws;
    float* partial = ws;                                           // 32*16*768
    float* xsum    = partial + (size_t)B_DIM * NSEG * D_DIM;       // 32*768
    float* invx    = xsum + (size_t)B_DIM * D_DIM;                 // 32
    float* invp    = invx + B_DIM;                                 // 1024
    int*   idxi    = (int*)(invp + POOL);                          // 32 ints

    k_copy_partial<<<B_DIM * NSEG, 192, 0, stream>>>(xe, pe, partial);
    k_xsum        <<<B_DIM,        192, 0, stream>>>(partial, xsum, invx);
    k_invp        <<<POOL,         192, 0, stream>>>(pk, invp);
    k_sim_wmma    <<<128,          32,  0, stream>>>(xsum, pk, invx, invp, sim);
    k_topk        <<<B_DIM,        32,  0, stream>>>(sim, tks, tki, idf, idxi);
    k_gather      <<<120,          256, 0, stream>>>(prompt, idxi, pe);
}